// TensorConvLayer_37134287242018
// MI455X (gfx1250) — compile-verified
//
#include <hip/hip_runtime.h>

typedef float v2f __attribute__((ext_vector_type(2)));
typedef float v8f __attribute__((ext_vector_type(8)));

#define N_NODES 10000
#define N_EDGES 160000
#define N_TILES (N_EDGES / 16)

#define INV_M0  0.25f           /* 1/sqrt(16) */
#define INV_3M1 0.20412414523f  /* 1/sqrt(24) */
#define INV_M1  0.35355339059f  /* 1/sqrt(8)  */
#define EPSV    1e-5f

/* workspace layout (float offsets) */
#define OFF_SUMS  0        /* 10000 x 40          */
#define OFF_CNTS  400000   /* 10000               */
#define OFF_STATS 410000   /* 40                  */
#define ZERO_N    410040
#define OFF_W1F   410048   /* 64 frags *32*2 = 4096   */
#define OFF_W2F   414144   /* 576 frags *32*2 = 36864 */

static __device__ __forceinline__ v8f wmma4(v2f a, v2f b, v8f c) {
  return __builtin_amdgcn_wmma_f32_16x16x4_f32(false, a, false, b, (short)0, c,
                                               false, false);
}

static __device__ __forceinline__ void atomAddF(float* p, float v) {
  __hip_atomic_fetch_add(p, v, __ATOMIC_RELAXED, __HIP_MEMORY_SCOPE_AGENT);
}

__global__ void zero_kernel(float* ws) {
  int i = blockIdx.x * blockDim.x + threadIdx.x;
  if (i < ZERO_N) ws[i] = 0.0f;
}

/* Repack W1 (64x64) and W2 (64x576) into WMMA B-fragment order:
   frag f = (kblock b, ntile j); lane = half*16 + n; 2 floats per lane:
   W[4b + 2*half + v][16j + n].  One contiguous float2 per lane per frag. */
__global__ void repack_kernel(const float* __restrict__ w1,
                              const float* __restrict__ w2,
                              float* __restrict__ w1f,
                              float* __restrict__ w2f) {
  int t = blockIdx.x * blockDim.x + threadIdx.x;
  int f = t >> 5, lane = t & 31;
  int hb = lane >> 4, nn = lane & 15;
  if (f < 64) {                       /* W1: b=f>>2, j=f&3 */
    int b = f >> 2, j = f & 3;
    int r0 = 4 * b + 2 * hb;
    w1f[(f * 32 + lane) * 2 + 0] = w1[(r0 + 0) * 64 + 16 * j + nn];
    w1f[(f * 32 + lane) * 2 + 1] = w1[(r0 + 1) * 64 + 16 * j + nn];
  } else if (f < 64 + 576) {          /* W2: b=f2/36, j=f2%36 */
    int f2 = f - 64;
    int b = f2 / 36, j = f2 % 36;
    int r0 = 4 * b + 2 * hb;
    w2f[(f2 * 32 + lane) * 2 + 0] = w2[(r0 + 0) * 576 + 16 * j + nn];
    w2f[(f2 * 32 + lane) * 2 + 1] = w2[(r0 + 1) * 576 + 16 * j + nn];
  }
}

#define WPB 6        /* waves per block */
#define SMW 2176     /* floats of LDS per wave */

__global__ __launch_bounds__(WPB * 32) void edge_kernel(
    const float* __restrict__ atom, const float* __restrict__ ef,
    const float* __restrict__ esh, const int* __restrict__ eidx,
    const float* __restrict__ b1, const float* __restrict__ b2,
    const float* __restrict__ w1f, const float* __restrict__ w2f,
    float* __restrict__ sums, float* __restrict__ cnts) {
  __shared__ float smem[WPB * SMW];
  __shared__ int nodes[WPB][16];

  int lane = threadIdx.x & 31;
  int wave = threadIdx.x >> 5;
  int tile = blockIdx.x * WPB + wave;
  if (tile >= N_TILES) return;
  int e0 = tile * 16;
  int hb = lane >> 4;    /* wave half */
  int nn = lane & 15;    /* column within 16-wide tile / edge row for A */

  float* sm  = smem + wave * SMW;
  float* hT  = sm;             /* 16 rows x stride 68 = 1088 */
  float* a11 = sm + 1088;      /* [e*16+j]  x0*sh0/4       */
  float* a12 = sm + 1344;      /* [e*16+i]  x0/4           */
  float* a13 = sm + 1600;      /* [e*24+i*3+m] sh0*x1/sqrt8 */
  float* a14 = sm + 1984;      /* [e*8+i]   (x1.sh1)/sqrt24 */
  float* s1  = sm + 2112;      /* [e*4+m]   sh1            */

  /* ---- per-edge multiplier tables (lanes 0..15, one edge each) ---- */
  if (lane < 16) {
    int e = e0 + lane;
    nodes[wave][lane] = eidx[N_EDGES + e];          /* edge_src */
    float4 s = *(const float4*)(esh + e * 4);
    const float* xr = atom + (size_t)eidx[e] * 40;  /* gather by edge_dst */
    float x[40];
#pragma unroll
    for (int t = 0; t < 10; t++) {
      float4 v = *(const float4*)(xr + t * 4);
      x[t * 4 + 0] = v.x; x[t * 4 + 1] = v.y;
      x[t * 4 + 2] = v.z; x[t * 4 + 3] = v.w;
    }
    float sh0 = s.x;
    s1[lane * 4 + 0] = s.y;
    s1[lane * 4 + 1] = s.z;
    s1[lane * 4 + 2] = s.w;
#pragma unroll
    for (int j = 0; j < 16; j++) {
      a11[lane * 16 + j] = x[j] * sh0 * INV_M0;
      a12[lane * 16 + j] = x[j] * INV_M0;
    }
#pragma unroll
    for (int i = 0; i < 8; i++) {
      float m0 = x[16 + i * 3 + 0];
      float m1 = x[16 + i * 3 + 1];
      float m2 = x[16 + i * 3 + 2];
      a13[lane * 24 + i * 3 + 0] = sh0 * m0 * INV_M1;
      a13[lane * 24 + i * 3 + 1] = sh0 * m1 * INV_M1;
      a13[lane * 24 + i * 3 + 2] = sh0 * m2 * INV_M1;
      a14[lane * 8 + i] = (m0 * s.y + m1 * s.z + m2 * s.w) * INV_3M1;
    }
  }
  asm volatile("s_wait_dscnt 0" ::: "memory");

  /* ---- GEMM1: h = relu(EF(16x64) @ W1(64x64) + b1), via 4x16 WMMAs ---- */
  v2f a[16];
  const float* efr = ef + (size_t)(e0 + nn) * 64 + 2 * hb;
#pragma unroll
  for (int b = 0; b < 16; b++) a[b] = *(const v2f*)(efr + 4 * b);

#pragma unroll
  for (int j = 0; j < 4; j++) {
    v8f c;
    float bias = b1[16 * j + nn];
#pragma unroll
    for (int r = 0; r < 8; r++) c[r] = bias;
#pragma unroll
    for (int b = 0; b < 16; b++) {
      v2f wb = *(const v2f*)(w1f + ((b * 4 + j) * 32 + lane) * 2);
      c = wmma4(a[b], wb, c);
    }
#pragma unroll
    for (int r = 0; r < 8; r++) {
      float v = c[r] > 0.0f ? c[r] : 0.0f;          /* relu */
      hT[(r + 8 * hb) * 68 + 16 * j + nn] = v;      /* C-layout -> LDS */
    }
  }
  asm volatile("s_wait_dscnt 0" ::: "memory");

  /* reload h as WMMA A-fragments (transpose through LDS) */
  v2f ha[16];
#pragma unroll
  for (int b = 0; b < 16; b++)
    ha[b] = *(const v2f*)(hT + nn * 68 + 4 * b + 2 * hb);

  /* ---- GEMM2 (h @ W2 + b2) with fused tensor product ---- */
  float o0[8] = {0, 0, 0, 0, 0, 0, 0, 0};
  float p1[8] = {0, 0, 0, 0, 0, 0, 0, 0};
  float q0[8] = {0, 0, 0, 0, 0, 0, 0, 0};
  float q1[8] = {0, 0, 0, 0, 0, 0, 0, 0};
  float q2[8] = {0, 0, 0, 0, 0, 0, 0, 0};

  /* w11 block: ntile j2 = row i of 16x16 -> rank-1 into out0 */
  for (int j2 = 0; j2 < 16; j2++) {
    v8f c;
    float bias = b2[16 * j2 + nn];
#pragma unroll
    for (int r = 0; r < 8; r++) c[r] = bias;
#pragma unroll
    for (int b = 0; b < 16; b++) {
      v2f wb = *(const v2f*)(w2f + ((b * 36 + j2) * 32 + lane) * 2);
      c = wmma4(ha[b], wb, c);
    }
#pragma unroll
    for (int r = 0; r < 8; r++)
      o0[r] += c[r] * a11[(r + 8 * hb) * 16 + j2];
  }
  /* w12 block: two rows per ntile, split across nn<8 / nn>=8 */
  for (int j2 = 16; j2 < 24; j2++) {
    v8f c;
    float bias = b2[16 * j2 + nn];
#pragma unroll
    for (int r = 0; r < 8; r++) c[r] = bias;
#pragma unroll
    for (int b = 0; b < 16; b++) {
      v2f wb = *(const v2f*)(w2f + ((b * 36 + j2) * 32 + lane) * 2);
      c = wmma4(ha[b], wb, c);
    }
    int i = 2 * (j2 - 16) + (nn >> 3);
#pragma unroll
    for (int r = 0; r < 8; r++)
      p1[r] += c[r] * a12[(r + 8 * hb) * 16 + i];
  }
  /* w13 block: per-m accumulators */
  for (int j2 = 24; j2 < 28; j2++) {
    v8f c;
    float bias = b2[16 * j2 + nn];
#pragma unroll
    for (int r = 0; r < 8; r++) c[r] = bias;
#pragma unroll
    for (int b = 0; b < 16; b++) {
      v2f wb = *(const v2f*)(w2f + ((b * 36 + j2) * 32 + lane) * 2);
      c = wmma4(ha[b], wb, c);
    }
    int i = 2 * (j2 - 24) + (nn >> 3);
#pragma unroll
    for (int r = 0; r < 8; r++) {
      float cv = c[r];
      int base = (r + 8 * hb) * 24 + i * 3;
      q0[r] += cv * a13[base + 0];
      q1[r] += cv * a13[base + 1];
      q2[r] += cv * a13[base + 2];
    }
  }
  /* w14 block: rank-1 into out0 */
  for (int j2 = 28; j2 < 36; j2++) {
    v8f c;
    float bias = b2[16 * j2 + nn];
#pragma unroll
    for (int r = 0; r < 8; r++) c[r] = bias;
#pragma unroll
    for (int b = 0; b < 16; b++) {
      v2f wb = *(const v2f*)(w2f + ((b * 36 + j2) * 32 + lane) * 2);
      c = wmma4(ha[b], wb, c);
    }
    int i = j2 - 28;
#pragma unroll
    for (int r = 0; r < 8; r++)
      o0[r] += c[r] * a14[(r + 8 * hb) * 8 + i];
  }

  /* merge even/odd row halves (lanes nn and nn^8 hold partials) */
#pragma unroll
  for (int r = 0; r < 8; r++) {
    p1[r] += __shfl_xor(p1[r], 8, 32);
    q0[r] += __shfl_xor(q0[r], 8, 32);
    q1[r] += __shfl_xor(q1[r], 8, 32);
    q2[r] += __shfl_xor(q2[r], 8, 32);
  }

  /* ---- scatter-add into node sums ---- */
#pragma unroll
  for (int r = 0; r < 8; r++) {
    int e = r + 8 * hb;
    atomAddF(sums + (size_t)nodes[wave][e] * 40 + nn, o0[r]);
  }
  if (nn < 8) {
#pragma unroll
    for (int r = 0; r < 8; r++) {
      int e = r + 8 * hb;
      float* dst = sums + (size_t)nodes[wave][e] * 40 + 16 + nn * 3;
      float sy = s1[e * 4 + 0], sz = s1[e * 4 + 1], sw = s1[e * 4 + 2];
      atomAddF(dst + 0, p1[r] * sy + q0[r]);
      atomAddF(dst + 1, p1[r] * sz + q1[r]);
      atomAddF(dst + 2, p1[r] * sw + q2[r]);
    }
  }
  if (lane < 16) atomAddF(cnts + nodes[wave][lane], 1.0f);
}

/* mean + residual, hierarchical BN-stat reduction */
__global__ void node_stage1(const float* __restrict__ atom, float* ws) {
  int c = threadIdx.x;   /* 0..39 */
  int ty = threadIdx.y;  /* 0..5  */
  float* sums = ws + OFF_SUMS;
  const float* cnts = ws + OFF_CNTS;
  float* stats = ws + OFF_STATS;
  float sA = 0.0f, sB = 0.0f;
  for (int n = blockIdx.x * 6 + ty; n < N_NODES; n += gridDim.x * 6) {
    float cnt = cnts[n];
    float inv = 1.0f / fmaxf(cnt, 1.0f);
    float y = sums[(size_t)n * 40 + c] * inv + atom[(size_t)n * 40 + c];
    sums[(size_t)n * 40 + c] = y;  /* reuse sums as y buffer */
    sA += y;
    sB += y * y;
  }
  __shared__ float rA[6][40], rB[6][40];
  rA[ty][c] = sA;
  rB[ty][c] = sB;
  __syncthreads();
  if (ty == 0) {
    float tA = 0.0f, tB = 0.0f;
#pragma unroll
    for (int k = 0; k < 6; k++) { tA += rA[k][c]; tB += rB[k][c]; }
    if (c < 16) {
      atomAddF(stats + c, tA);
      atomAddF(stats + 16 + c, tB);
    } else {
      atomAddF(stats + 32 + (c - 16) / 3, tB);
    }
  }
}

/* apply irreps batch norm */
__global__ void node_stage2(const float* __restrict__ ws,
                            const float* __restrict__ bnw,
                            const float* __restrict__ bnb,
                            float* __restrict__ out) {
  int idx = blockIdx.x * blockDim.x + threadIdx.x;
  if (idx >= N_NODES * 40) return;
  int c = idx % 40;
  const float* y = ws + OFF_SUMS;
  const float* stats = ws + OFF_STATS;
  float v = y[idx];
  float o;
  if (c < 16) {
    float mean = stats[c] * (1.0f / N_NODES);
    float ex2 = stats[16 + c] * (1.0f / N_NODES);
    float var = ex2 - mean * mean;
    o = (v - mean) * rsqrtf(var + EPSV) * bnw[c] + bnb[c];
  } else {
    int j = (c - 16) / 3;
    float vn = stats[32 + j] * (1.0f / (3.0f * N_NODES));
    o = v * rsqrtf(vn + EPSV) * bnw[16 + j];
  }
  out[idx] = o;
}

extern "C" void kernel_launch(void* const* d_in, const int* in_sizes, int n_in,
                              void* d_out, int out_size, void* d_ws,
                              size_t ws_size, hipStream_t stream) {
  (void)in_sizes; (void)n_in; (void)out_size; (void)ws_size;
  const float* atom = (const float*)d_in[0];
  const float* ef   = (const float*)d_in[1];
  const float* esh  = (const float*)d_in[2];
  const int*   eidx = (const int*)d_in[3];
  const float* w1   = (const float*)d_in[4];
  const float* b1   = (const float*)d_in[5];
  const float* w2   = (const float*)d_in[6];
  const float* b2   = (const float*)d_in[7];
  const float* bnw  = (const float*)d_in[8];
  const float* bnb  = (const float*)d_in[9];
  float* ws  = (float*)d_ws;
  float* out = (float*)d_out;

  zero_kernel<<<(ZERO_N + 255) / 256, 256, 0, stream>>>(ws);
  repack_kernel<<<(640 * 32 + 255) / 256, 256, 0, stream>>>(
      w1, w2, ws + OFF_W1F, ws + OFF_W2F);
  edge_kernel<<<(N_TILES + WPB - 1) / WPB, WPB * 32, 0, stream>>>(
      atom, ef, esh, eidx, b1, b2, ws + OFF_W1F, ws + OFF_W2F,
      ws + OFF_SUMS, ws + OFF_CNTS);
  node_stage1<<<128, dim3(40, 6), 0, stream>>>(atom, ws);
  node_stage2<<<(N_NODES * 40 + 255) / 256, 256, 0, stream>>>(ws, bnw, bnb, out);
}